// BayesRiskCTC_11278584119725
// MI455X (gfx1250) — compile-verified
//
#include <hip/hip_runtime.h>
#include <hip/hip_bf16.h>

// Bayes-Risk CTC on MI455X (gfx1250).
//
// Latency-bound serial scan over T=1000; CDNA5-specific path used here is the
// async gather-to-LDS pipe (GLOBAL_LOAD_ASYNC_TO_LDS_B32 + s_wait_asynccnt)
// to double-buffer the per-timestep gathered probability row p_ext[b,t,:]
// (201 scattered floats from a 500-float row) while the wave executes the
// dependent log-sum-exp recurrence. WMMA is inapplicable (no matmul).

#define B_ 32
#define T_ 1000
#define C_ 500
#define U_ 100
#define S_ 201              // 2U+1
#define RISK_F 0.1f
#define NEG_INF (-__builtin_inff())

#ifndef __has_builtin
#define __has_builtin(x) 0
#endif

#if __has_builtin(__builtin_amdgcn_global_load_async_to_lds_b32)
#define HAS_ASYNC_LDS 1
#else
#define HAS_ASYNC_LDS 0
#endif

#if __has_builtin(__builtin_amdgcn_s_wait_asynccnt)
#define WAIT_ASYNC(n) __builtin_amdgcn_s_wait_asynccnt(n)
#elif HAS_ASYNC_LDS
#define WAIT_ASYNC(n) asm volatile("s_wait_asynccnt %0" ::"n"(n) : "memory")
#else
#define WAIT_ASYNC(n) do {} while (0)
#endif

// Per-lane gather: copies one float from global gsrc to LDS at ldsdst, asynchronously (ASYNCcnt).
__device__ __forceinline__ void gather_to_lds(const float* gsrc, float* ldsdst) {
#if HAS_ASYNC_LDS
  __builtin_amdgcn_global_load_async_to_lds_b32(
      (__attribute__((address_space(1))) int*)(void*)gsrc,
      (__attribute__((address_space(3))) int*)(void*)ldsdst,
      0, 0);
#else
  *ldsdst = *gsrc;
#endif
}

__device__ __forceinline__ bool is_neg_inf(float x) { return x == NEG_INF; }
__device__ __forceinline__ bool is_inf_any(float x) { return fabsf(x) == __builtin_inff(); }

// Matches reference _lse3 exactly (term order preserved).
__device__ __forceinline__ float lse3(float x1, float x2, float x3) {
  float m = fmaxf(fmaxf(x1, x2), x3);
  bool dead = is_neg_inf(m);
  float ms = dead ? 0.0f : m;
  float s = expf(x1 - ms) + expf(x2 - ms) + expf(x3 - ms);
  s = dead ? 1.0f : s;
  return dead ? NEG_INF : (ms + logf(s));
}

// Matches reference _log_sub_exp exactly (incl. the -2000/-2001 fallback).
__device__ __forceinline__ float log_sub_exp(float a, float b) {
  bool mask1 = !is_inf_any(a) && !is_inf_any(b);
  float a_ = mask1 ? a : -1.0f;
  float b_ = mask1 ? b : -2.0f;
  float ans_tmp = b_ + logf(expf(a_ - b_) - 1.0f);
  bool bad = is_inf_any(ans_tmp);
  float a2 = bad ? -2000.0f : a_;
  float b2 = bad ? -2001.0f : b_;
  float ans = b2 + logf(expf(a2 - b2) - 1.0f);
  ans = mask1 ? ans : NEG_INF;
  bool mask2 = !is_inf_any(a) && is_inf_any(b);
  return mask2 ? a : ans;
}

// ---------------------------------------------------------------------------
// Pass 1: backward beta recursion; writes beta_prime[b][u][t] to ws.
// beta_prime[u,t] = log_sub_exp(beta[t][2u+1], p[t+1][2u+1] + beta[t+1][2u+1])
// (uniform formula; t=T-1 uses synthetic -inf q, giving b_l[:, :, -1]).
// ---------------------------------------------------------------------------
__global__ __launch_bounds__(256) void brctc_backward(
    const float* __restrict__ nnet, const int* __restrict__ ys,
    const int* __restrict__ hlens, const int* __restrict__ ylens,
    float* __restrict__ bp) {
  const int b = blockIdx.x;
  const int s = threadIdx.x;
  const int hlen = hlens[b];
  const int ylen = ylens[b];
  const int* ysb = ys + b * U_;

  __shared__ float sh_beta[S_];
  __shared__ float sh_q[S_ + 2];
  __shared__ float sh_p[2][S_];

  int lab = 0;
  bool skip2 = false;   // allow_skip at s+2 (shift-left by 2)
  bool valid = false;   // s <= 2*ylen
  float bfin = NEG_INF;
  if (s < S_) {
    valid = (s <= 2 * ylen);
    if (s & 1) lab = ysb[(s - 1) >> 1];
    int s2 = s + 2;
    if (s2 < S_ && (s2 & 1) && s2 >= 3)
      skip2 = (ysb[(s2 - 1) >> 1] != ysb[(s2 - 3) >> 1]);
    bfin = (s == 2 * ylen || s == 2 * ylen - 1) ? 0.0f : NEG_INF;
    sh_beta[s] = NEG_INF;           // reverse-scan init carry
  }
  if (s < 2) sh_q[S_ + s] = NEG_INF;  // shift-left padding

  const float* nb = nnet + (size_t)b * T_ * C_;

  for (int t = T_ - 1; t >= 0; --t) {
    // Prefetch row p[t] (consumed at iteration t-1). Slot parity t&1.
    if (t >= 1 && s < S_)
      gather_to_lds(nb + (size_t)t * C_ + lab, &sh_p[t & 1][s]);
    // Retire row p[t+1] (issued one iteration ago; async loads complete in order).
    if (t >= 1) { WAIT_ASYNC(1); } else { WAIT_ASYNC(0); }
    __syncthreads();  // publish row t+1 + previous sh_beta to all waves

    if (s < S_) {
      float qv;
      if (t == T_ - 1) {
        qv = NEG_INF;  // p_next row is -inf by construction
      } else {
        float pv = valid ? sh_p[(t + 1) & 1][s] : NEG_INF;
        qv = pv + sh_beta[s];  // -inf absorbs correctly
      }
      sh_q[s] = qv;
    }
    __syncthreads();  // q ready; all reads of sh_beta/sh_p[t+1] done

    if (s < S_) {
      float q0 = sh_q[s];
      float q1 = sh_q[s + 1];
      float q2 = skip2 ? sh_q[s + 2] : NEG_INF;
      float cand = lse3(q0, q1, q2);
      float bnew = (t == hlen - 1) ? bfin : cand;
      if (s & 1) {
        int u = (s - 1) >> 1;
        bp[((size_t)b * U_ + u) * T_ + t] = log_sub_exp(bnew, q0);
      }
      sh_beta[s] = bnew;  // safe: readers of sh_beta are past barrier above
    }
  }
}

// ---------------------------------------------------------------------------
// Pass 2: forward alpha recursion fused with
//   loss_state[b][u][t] = (t < hlen ? alpha[t][2u+1] : -inf)
//                         + beta_prime[b][u][t] + risk(t)
// overwriting ws in place.
// ---------------------------------------------------------------------------
__global__ __launch_bounds__(256) void brctc_forward(
    const float* __restrict__ nnet, const int* __restrict__ ys,
    const int* __restrict__ hlens, const int* __restrict__ ylens,
    float* __restrict__ bp_ls) {
  const int b = blockIdx.x;
  const int s = threadIdx.x;
  const int hlen = hlens[b];
  const int ylen = ylens[b];
  const int* ysb = ys + b * U_;

  __shared__ float sh_a_raw[S_ + 2];  // [0..1] = -inf pads for s-1, s-2
  __shared__ float sh_p[2][S_];
  float* sh_a = sh_a_raw + 2;

  int lab = 0;
  bool skip = false;
  bool valid = false;
  if (s < S_) {
    valid = (s <= 2 * ylen);
    if (s & 1) lab = ysb[(s - 1) >> 1];
    if ((s & 1) && s >= 3)
      skip = (ysb[(s - 1) >> 1] != ysb[(s - 3) >> 1]);
    sh_a[s] = (s == 0) ? 0.0f : NEG_INF;  // a0
  }
  if (s < 2) sh_a_raw[s] = NEG_INF;

  const float* nb = nnet + (size_t)b * T_ * C_;
  const float hf = (float)hlen;

  // Prologue: prefetch row p[0].
  if (s < S_) gather_to_lds(nb + lab, &sh_p[0][s]);

  for (int t = 0; t < T_; ++t) {
    if (t + 1 < T_ && s < S_)
      gather_to_lds(nb + (size_t)(t + 1) * C_ + lab, &sh_p[(t + 1) & 1][s]);
    if (t + 1 < T_) { WAIT_ASYNC(1); } else { WAIT_ASYNC(0); }
    __syncthreads();  // row t visible; sh_a from previous step visible

    float anew = NEG_INF;
    if (s < S_) {
      float pv = valid ? sh_p[t & 1][s] : NEG_INF;
      float a0 = sh_a[s];
      float a1 = sh_a[s - 1];
      float a2 = skip ? sh_a[s - 2] : NEG_INF;
      anew = pv + lse3(a0, a1, a2);
    }
    __syncthreads();  // all neighbor reads of sh_a complete before overwrite

    if (s < S_) {
      sh_a[s] = anew;
      if (s & 1) {
        int u = (s - 1) >> 1;
        size_t idx = ((size_t)b * U_ + u) * T_ + t;
        float a_l = (t < hlen) ? anew : NEG_INF;
        float risk = -RISK_F * ((float)t / hf);
        float ls = (a_l + bp_ls[idx]) + risk;  // reference add order
        bp_ls[idx] = ls;
      }
    }
  }
}

// ---------------------------------------------------------------------------
// Pass 3: exact two-pass safe-LSE over t per (b,u); then count non(-inf)
// entries, pick loss_u[count-1], negate.
// ---------------------------------------------------------------------------
__global__ __launch_bounds__(128) void brctc_reduce(
    const float* __restrict__ ls, float* __restrict__ out) {
  const int b = blockIdx.x;
  const int u = threadIdx.x;
  __shared__ float lu_sh[U_];

  if (u < U_) {
    const float* row = ls + ((size_t)b * U_ + u) * T_;
    float m = NEG_INF;
    for (int t = 0; t < T_; ++t) m = fmaxf(m, row[t]);
    float lu;
    if (is_neg_inf(m)) {
      lu = NEG_INF;
    } else {
      float acc = 0.0f;
      for (int t = 0; t < T_; ++t) acc += expf(row[t] - m);
      lu = m + logf(acc);
    }
    lu_sh[u] = lu;
  }
  __syncthreads();

  if (u == 0) {
    int cnt = 0;
    for (int i = 0; i < U_; ++i) cnt += is_neg_inf(lu_sh[i]) ? 0 : 1;
    int last = cnt - 1;
    last = last < 0 ? 0 : (last > U_ - 1 ? U_ - 1 : last);  // JAX clip mode
    out[b] = -lu_sh[last];
  }
}

extern "C" void kernel_launch(void* const* d_in, const int* in_sizes, int n_in,
                              void* d_out, int out_size, void* d_ws, size_t ws_size,
                              hipStream_t stream) {
  const float* nnet = (const float*)d_in[0];  // [B,T,C] f32 (log-softmax)
  const int* ys     = (const int*)d_in[1];    // [B,U]
  const int* hl     = (const int*)d_in[2];    // [B]
  const int* yl     = (const int*)d_in[3];    // [B]
  float* ws         = (float*)d_ws;           // B*U*T floats = 12.8 MB

  brctc_backward<<<B_, 256, 0, stream>>>(nnet, ys, hl, yl, ws);
  brctc_forward <<<B_, 256, 0, stream>>>(nnet, ys, hl, yl, ws);
  brctc_reduce  <<<B_, 128, 0, stream>>>(ws, (float*)d_out);
}